// AttentionLayer_54391465836888
// MI455X (gfx1250) — compile-verified
//
#include <hip/hip_runtime.h>
#include <math.h>

// Problem constants (match reference)
#define Bn 4
#define TE 512
#define TD 256
#define HE 128
#define HD 256

typedef float v2f __attribute__((ext_vector_type(2)));
typedef float v8f __attribute__((ext_vector_type(8)));

__device__ __forceinline__ float fast_tanh(float x) {
#if __has_builtin(__builtin_amdgcn_tanhf)
    return __builtin_amdgcn_tanhf(x);
#else
    return tanhf(x);
#endif
}

// One wave computes one 16x16 fp32 tile: D[16,16] = A[16,K] * B[K,16]
// A row-major with leading dim lda, B row-major with leading dim ldb.
// A-frag layout (V_WMMA_F32_16X16X4_F32): lanes 0-15 hold K=k0,k0+1 in v[0],v[1];
// lanes 16-31 hold K=k0+2,k0+3. B-frag is the mirror (lane = N).
__device__ __forceinline__ void gemm_tile_f32(const float* __restrict__ A, int lda,
                                              const float* __restrict__ Bm, int ldb,
                                              float* __restrict__ D, int ldd,
                                              int K, int lane) {
    const int mn   = lane & 15;
    const int half = lane >> 4;
    v8f acc = {};
    for (int k0 = 0; k0 < K; k0 += 4) {
        const int ka = k0 + half * 2;
        v2f a, b;
        a.x = A[mn * lda + ka];
        a.y = A[mn * lda + ka + 1];
        b.x = Bm[ka * ldb + mn];
        b.y = Bm[(ka + 1) * ldb + mn];
        acc = __builtin_amdgcn_wmma_f32_16x16x4_f32(
            false, a, false, b, (short)0, acc, false, false);
    }
#pragma unroll
    for (int r = 0; r < 8; ++r)
        D[(r + 8 * half) * ldd + mn] = acc[r];
}

// Kernel 1: Ws = enc @ W_a  [Bn,TE,HE], Uh = dec @ U_a  [Bn,TD,HE]
// One wave per 16x16 tile; 8 waves per block.
#define WS_TILES (Bn * (TE / 16) * (HE / 16))   // 1024
#define UH_TILES (Bn * (TD / 16) * (HE / 16))   // 512
#define PROJ_BLOCKS ((WS_TILES + UH_TILES) / 8) // 192

__global__ __launch_bounds__(256)
void proj_kernel(const float* __restrict__ enc, const float* __restrict__ dec,
                 const float* __restrict__ W,   const float* __restrict__ U,
                 float* __restrict__ Ws,        float* __restrict__ Uh) {
    const int lane = threadIdx.x & 31;
    const int tile = blockIdx.x * 8 + (threadIdx.x >> 5);
    if (tile < WS_TILES) {
        // enc[b] (TE x HE) @ W (HE x HE) -> Ws[b] (TE x HE)
        const int b   = tile >> 8;          // 256 tiles per batch
        const int rem = tile & 255;
        const int i   = rem >> 3;           // M tile (32)
        const int j   = rem & 7;            // N tile (8)
        gemm_tile_f32(enc + (b * TE + i * 16) * HE, HE,
                      W + j * 16, HE,
                      Ws + (b * TE + i * 16) * HE + j * 16, HE,
                      HE, lane);
    } else {
        // dec[b] (TD x HD) @ U (HD x HE) -> Uh[b] (TD x HE)
        const int t2  = tile - WS_TILES;
        const int b   = t2 >> 7;            // 128 tiles per batch
        const int rem = t2 & 127;
        const int i   = rem >> 3;
        const int j   = rem & 7;
        gemm_tile_f32(dec + (b * TD + i * 16) * HD, HD,
                      U + j * 16, HE,
                      Uh + (b * TD + i * 16) * HE + j * 16, HE,
                      HD, lane);
    }
}

// Kernel 2: fused score (tanh dot with V) + softmax + context GEMM.
// One block = one (batch, 16 decoder steps). 256 threads = 8 waves.
__global__ __launch_bounds__(256)
void attn_kernel(const float* __restrict__ enc, const float* __restrict__ Ws,
                 const float* __restrict__ Uh,  const float* __restrict__ V,
                 float* __restrict__ c_out,     float* __restrict__ e_out) {
    __shared__ float sUh[16][HE + 4];     // decoder-projection tile
    __shared__ float sV[HE];
    __shared__ float sAlpha[16][TE + 4];  // scores -> softmax weights (padded: no bank conflict)

    const int tid  = threadIdx.x;
    const int b    = blockIdx.x >> 4;     // TD/16 = 16 tiles per batch
    const int t0   = (blockIdx.x & 15) * 16;

    // Stage Uh tile and V into LDS
    for (int idx = tid; idx < 16 * HE; idx += 256) {
        const int t = idx >> 7, f = idx & (HE - 1);
        sUh[t][f] = Uh[(b * TD + t0 + t) * HE + f];
    }
    for (int idx = tid; idx < HE; idx += 256) sV[idx] = V[idx];
    __syncthreads();

    // Phase 1: scores e[t][s] = sum_f V[f] * tanh(Ws[b,s,f] + Uh[t,f]).
    // 16*512 = 8192 dot products; TRANS-bound (67M tanh over the grid).
    const float* __restrict__ WsB = Ws + b * TE * HE;
    for (int p = tid; p < 16 * TE; p += 256) {
        const int t = p >> 9;             // TE = 512
        const int s = p & (TE - 1);
        const float4* __restrict__ wrow = (const float4*)(WsB + s * HE);
        const float4* __restrict__ urow = (const float4*)(&sUh[t][0]);
        const float4* __restrict__ vrow = (const float4*)(&sV[0]);
        float acc = 0.f;
#pragma unroll 8
        for (int f4 = 0; f4 < HE / 4; ++f4) {
            const float4 w = wrow[f4];
            const float4 u = urow[f4];
            const float4 v = vrow[f4];
            acc += v.x * fast_tanh(w.x + u.x);
            acc += v.y * fast_tanh(w.y + u.y);
            acc += v.z * fast_tanh(w.z + u.z);
            acc += v.w * fast_tanh(w.w + u.w);
        }
        sAlpha[t][s] = acc;
    }
    __syncthreads();

    // Phase 2: row softmax over s (512). One wave per row, 2 rows per wave.
    const int wave = tid >> 5;
    const int lane = tid & 31;
    for (int t = wave; t < 16; t += 8) {
        float mx = -INFINITY;
        for (int s = lane; s < TE; s += 32) mx = fmaxf(mx, sAlpha[t][s]);
#pragma unroll
        for (int off = 16; off > 0; off >>= 1) mx = fmaxf(mx, __shfl_xor(mx, off, 32));
        float sum = 0.f;
        for (int s = lane; s < TE; s += 32) {
            const float ex = __expf(sAlpha[t][s] - mx);
            sAlpha[t][s] = ex;
            sum += ex;
        }
#pragma unroll
        for (int off = 16; off > 0; off >>= 1) sum += __shfl_xor(sum, off, 32);
        const float inv = 1.0f / sum;
        for (int s = lane; s < TE; s += 32) {
            const float a = sAlpha[t][s] * inv;
            sAlpha[t][s] = a;
            e_out[(b * TD + t0 + t) * TE + s] = a;
        }
    }
    __syncthreads();

    // Phase 3: context c[16,HE] = alpha[16,TE] @ enc[b][TE,HE] via fp32 WMMA.
    // Wave w owns N-tile w (16 columns); K loop over TE in steps of 4.
    const float* __restrict__ encB = enc + b * TE * HE;
    const int mn   = lane & 15;
    const int half = lane >> 4;
    const int n0   = wave * 16;
    v8f acc = {};
    for (int k0 = 0; k0 < TE; k0 += 4) {
        const int ka = k0 + half * 2;
        v2f a, bf;
        a.x  = sAlpha[mn][ka];
        a.y  = sAlpha[mn][ka + 1];
        bf.x = encB[ka * HE + n0 + mn];
        bf.y = encB[(ka + 1) * HE + n0 + mn];
        acc = __builtin_amdgcn_wmma_f32_16x16x4_f32(
            false, a, false, bf, (short)0, acc, false, false);
    }
#pragma unroll
    for (int r = 0; r < 8; ++r)
        c_out[(b * TD + t0 + r + 8 * half) * HE + n0 + mn] = acc[r];
}

extern "C" void kernel_launch(void* const* d_in, const int* in_sizes, int n_in,
                              void* d_out, int out_size, void* d_ws, size_t ws_size,
                              hipStream_t stream) {
    const float* enc = (const float*)d_in[0]; // [B,TE,HE]
    const float* dec = (const float*)d_in[1]; // [B,TD,HD]
    const float* W   = (const float*)d_in[2]; // [HE,HE]
    const float* U   = (const float*)d_in[3]; // [HD,HE]
    const float* V   = (const float*)d_in[4]; // [HE,1] -> flat [HE]

    float* ws_Ws = (float*)d_ws;              // [B,TE,HE] = 1 MB
    float* ws_Uh = ws_Ws + Bn * TE * HE;      // [B,TD,HE] = 0.5 MB

    float* c_out = (float*)d_out;             // [B,TD,HE]
    float* e_out = c_out + Bn * TD * HE;      // [B,TD,TE]

    proj_kernel<<<PROJ_BLOCKS, 256, 0, stream>>>(enc, dec, W, U, ws_Ws, ws_Uh);
    attn_kernel<<<Bn * (TD / 16), 256, 0, stream>>>(enc, ws_Ws, ws_Uh, V, c_out, e_out);
}